// Transformer_60670708023310
// MI455X (gfx1250) — compile-verified
//
#include <hip/hip_runtime.h>
#include <cmath>

#define Bn 64
#define Sn 256
#define Dn 1024
#define Tn 60
#define HIST 20
#define D4 (Dn / 4)   // 256 float4 chunks per row

typedef __attribute__((ext_vector_type(2))) float v2f;
typedef __attribute__((ext_vector_type(4))) float v4f;
typedef __attribute__((ext_vector_type(8))) float v8f;

struct LagW { float w[HIST]; };

__device__ __forceinline__ float sigmoidf_(float x) { return 1.0f / (1.0f + __expf(-x)); }

__device__ __forceinline__ float dp4(v4f a, v4f b) {
    return a.x * b.x + a.y * b.y + a.z * b.z + a.w * b.w;
}

__device__ __forceinline__ float wave_reduce_add(float v) {
#pragma unroll
    for (int m = 16; m >= 1; m >>= 1) v += __shfl_xor(v, m, 32);
    return v;
}

// ---- once: mem = e * add0[s]; add_state = add0[s] ----
__global__ void k_init(const v4f* __restrict__ e4, v4f* __restrict__ mem4,
                       float* __restrict__ ast) {
    long i = (long)blockIdx.x * blockDim.x + threadIdx.x; // B*S*D/4 threads
    int c = (int)(i % D4);
    long row = i / D4;
    int s = (int)(row % Sn);
    float a0 = (s < 50) ? (1.0f - (s + 1) / 50.0f) : 0.0f;
    mem4[i] = e4[i] * a0;
    if (c == 0) ast[row] = a0;
}

// ---- once: en1[b,s] = e_row . Wn[:D]; en2[b,s] = e_row . Wn[D:] ----
__global__ void k_en(const v4f* __restrict__ e4, const v4f* __restrict__ Wn4,
                     float* __restrict__ en1, float* __restrict__ en2) {
    int lane = threadIdx.x & 31, wave = threadIdx.x >> 5;
    int row = blockIdx.x * 8 + wave; // < B*S
    const v4f* er = e4 + (long)row * D4;
    float p1 = 0.f, p2 = 0.f;
#pragma unroll
    for (int k = 0; k < 8; k++) {
        int c = k * 32 + lane;
        v4f ev = er[c];
        p1 += dp4(ev, Wn4[c]);
        p2 += dp4(ev, Wn4[D4 + c]);
    }
    p1 = wave_reduce_add(p1);
    p2 = wave_reduce_add(p2);
    if (lane == 0) { en1[row] = p1; en2[row] = p2; }
}

// ---- per step: attn_w[b,s], lcn[b], remove/add gate scalars ----
__global__ void k_step_small(const float* __restrict__ attns, const float* __restrict__ queries,
                             const v4f* __restrict__ Wr4, const float* __restrict__ brp,
                             const v4f* __restrict__ Wa4, const float* __restrict__ bap,
                             const float* __restrict__ en2, LagW lw, int t, int jmax,
                             float* __restrict__ attn_w, float* __restrict__ rp,
                             float* __restrict__ ap, float* __restrict__ lcn) {
    int b = blockIdx.x;
    int tid = threadIdx.x; // 256 threads, one s each
    __shared__ float red[256];
    const float* ab = attns + (long)b * Tn * Sn + (long)t * Sn + tid;
    float aw = 0.f;
    for (int j = 0; j <= jmax; j++) aw += lw.w[j] * ab[-(long)j * Sn];
    attn_w[b * Sn + tid] = aw;
    // lcn[b] = sum_s attn_w * en2
    red[tid] = aw * en2[b * Sn + tid];
    __syncthreads();
    for (int o = 128; o > 0; o >>= 1) { if (tid < o) red[tid] += red[tid + o]; __syncthreads(); }
    float lsum = red[0]; __syncthreads();
    // remove/add gate dots over D (one float4 chunk per thread)
    const v4f* qv4 = (const v4f*)(queries + (long)b * Tn * Dn + (long)t * Dn);
    v4f qc = qv4[tid];
    float pr = dp4(qc, Wr4[tid]);
    float pa = dp4(qc, Wa4[tid]);
    red[tid] = pr; __syncthreads();
    for (int o = 128; o > 0; o >>= 1) { if (tid < o) red[tid] += red[tid + o]; __syncthreads(); }
    float prs = red[0]; __syncthreads();
    red[tid] = pa; __syncthreads();
    for (int o = 128; o > 0; o >>= 1) { if (tid < o) red[tid] += red[tid + o]; __syncthreads(); }
    if (tid == 0) {
        rp[b] = sigmoidf_(prs + brp[0]);
        ap[b] = sigmoidf_(red[0] + bap[0]);
        lcn[b] = lsum;
    }
}

// ---- per step: query_s[b,d] = sum_j w[j] * queries[b, t-j, d] ----
__global__ void k_query_s(const v4f* __restrict__ queries4, LagW lw, int t, int jmax,
                          v4f* __restrict__ qs4) {
    int idx = blockIdx.x * 256 + threadIdx.x; // B*D/4
    int b = idx >> 8, c = idx & 255;
    const v4f* qb = queries4 + (long)b * Tn * D4 + (long)t * D4 + c;
    v4f v = {};
    for (int j = 0; j <= jmax; j++) v += lw.w[j] * qb[-(long)j * D4];
    qs4[idx] = v;
}

// ---- per step: qWq = query_s(64x1024) @ Wq(1024x1024) via f32 WMMA ----
__global__ void k_gemm_wmma(const float* __restrict__ Qs, const float* __restrict__ Wq,
                            float* __restrict__ out) {
    int wave = threadIdx.x >> 5;
    int lane = threadIdx.x & 31;
    int tile = blockIdx.x * 4 + wave; // 256 tiles: 4 (M) x 64 (N)
    int tm = tile >> 6;
    int tn = tile & 63;
    int half = lane >> 4; // 0: K lo pair, 1: K hi pair (per 16x4 layout)
    int l16 = lane & 15;
    int arow = tm * 16 + l16;  // A: M = lane%16
    int bcol = tn * 16 + l16;  // B: N = lane%16
#if __has_builtin(__builtin_amdgcn_wmma_f32_16x16x4_f32)
    v8f acc = {};
    for (int k0 = 0; k0 < Dn; k0 += 8) {
        int ka = k0 + half * 2;
        v2f a0, b0, a1, b1;
        a0.x = Qs[arow * Dn + ka];
        a0.y = Qs[arow * Dn + ka + 1];
        b0.x = Wq[(long)ka * Dn + bcol];
        b0.y = Wq[(long)(ka + 1) * Dn + bcol];
        int kb = ka + 4;
        a1.x = Qs[arow * Dn + kb];
        a1.y = Qs[arow * Dn + kb + 1];
        b1.x = Wq[(long)kb * Dn + bcol];
        b1.y = Wq[(long)(kb + 1) * Dn + bcol];
        acc = __builtin_amdgcn_wmma_f32_16x16x4_f32(false, a0, false, b0,
                                                    (short)0, acc, false, false);
        acc = __builtin_amdgcn_wmma_f32_16x16x4_f32(false, a1, false, b1,
                                                    (short)0, acc, false, false);
    }
#pragma unroll
    for (int j = 0; j < 8; j++) {
        int m = tm * 16 + j + half * 8; // C/D layout: VGPR j -> M=j (lanes 0-15), M=j+8 (16-31)
        out[(long)m * Dn + tn * 16 + l16] = acc[j];
    }
#else
    // scalar fallback (compile-safety only)
    float acc[8] = {};
    for (int k = 0; k < Dn; k++) {
#pragma unroll
        for (int j = 0; j < 8; j++) {
            int m = tm * 16 + j + half * 8;
            acc[j] += Qs[m * Dn + k] * Wq[(long)k * Dn + bcol];
        }
    }
#pragma unroll
    for (int j = 0; j < 8; j++) {
        int m = tm * 16 + j + half * 8;
        out[(long)m * Dn + tn * 16 + l16] = acc[j];
    }
#endif
}

// ---- per step fused pass over mem/e: sim, sel partials, mem & add_state update ----
// All B*S*D traffic through 128-bit vector loads/stores (L2-resident working set).
__global__ void __launch_bounds__(256)
k_step_big(const float* __restrict__ attns, const v4f* __restrict__ e4,
           v4f* __restrict__ mem4, float* __restrict__ ast,
           const v4f* __restrict__ qWq4, const float* __restrict__ attn_w,
           const float* __restrict__ rp, const float* __restrict__ ap,
           const float* __restrict__ lcn, const float* __restrict__ en1,
           const float* __restrict__ bnp, int t, v4f* __restrict__ selpart4) {
    int chunk = blockIdx.x; // 0..3 (64 rows each)
    int b = blockIdx.y;     // 0..63
    int lane = threadIdx.x & 31;
    int wave = threadIdx.x >> 5; // 0..7
    __shared__ v4f ssel[8 * D4]; // 32 KB
    // per-wave copy of qWq[b,:] (reused across this wave's 8 rows)
    v4f qw[8];
    const v4f* qb = qWq4 + (long)b * D4;
#pragma unroll
    for (int k = 0; k < 8; k++) qw[k] = qb[k * 32 + lane];
    float rpb = rp[b], apb = ap[b], lcb = lcn[b], bnv = bnp[0];
    v4f sel[8];
#pragma unroll
    for (int k = 0; k < 8; k++) sel[k] = (v4f){};
    for (int r = 0; r < 8; r++) {
        int s = chunk * 64 + wave * 8 + r;
        long base = ((long)b * Sn + s) * D4;
        v4f m[8], ev[8];
#pragma unroll
        for (int k = 0; k < 8; k++) {
            m[k] = mem4[base + k * 32 + lane];
            ev[k] = e4[base + k * 32 + lane];
        }
        float dot = 0.f;
#pragma unroll
        for (int k = 0; k < 8; k++) dot += dp4(m[k], qw[k]);
        dot = wave_reduce_add(dot);
        float sim = sigmoidf_(dot);
        float aw = attn_w[b * Sn + s];
        float a0 = ast[b * Sn + s];
        float nxt = sigmoidf_(en1[b * Sn + s] + lcb + bnv);
        float ca = (1.0f - a0) * (apb * nxt);
        float rm = 1.0f - rpb * aw * sim;
        float at = attns[(long)b * Tn * Sn + (long)t * Sn + s];
        if (lane == 0) ast[b * Sn + s] = a0 + ca;
#pragma unroll
        for (int k = 0; k < 8; k++) {
            v4f mv = m[k];
            sel[k] += at * mv;                          // sel uses pre-update mem
            mem4[base + k * 32 + lane] = mv * rm + ev[k] * ca;
        }
    }
    // deterministic cross-wave sel reduction (128-bit LDS ops)
#pragma unroll
    for (int k = 0; k < 8; k++) ssel[wave * D4 + k * 32 + lane] = sel[k];
    __syncthreads();
    int tid = threadIdx.x; // 256 threads == D4 chunks
    v4f v = {};
#pragma unroll
    for (int w = 0; w < 8; w++) v += ssel[w * D4 + tid];
    selpart4[(((long)b * 4) + chunk) * D4 + tid] = v;
}

// ---- per step: reduce 4 chunk partials -> sels[b, t, :] ----
__global__ void k_sel_final(const v4f* __restrict__ selpart4, v4f* __restrict__ outp4, int t) {
    int b = blockIdx.x;
    int tid = threadIdx.x; // 256 == D4
    long p = (long)b * 4 * D4 + tid;
    v4f v = selpart4[p] + selpart4[p + D4] + selpart4[p + 2 * D4] + selpart4[p + 3 * D4];
    outp4[((long)b * Tn + t) * D4 + tid] = v;
}

extern "C" void kernel_launch(void* const* d_in, const int* in_sizes, int n_in,
                              void* d_out, int out_size, void* d_ws, size_t ws_size,
                              hipStream_t stream) {
    const float* e   = (const float*)d_in[0];
    const float* q   = (const float*)d_in[1];
    const float* at  = (const float*)d_in[2];
    const float* Wr  = (const float*)d_in[3];
    const float* br  = (const float*)d_in[4];
    const float* Wa  = (const float*)d_in[5];
    const float* ba  = (const float*)d_in[6];
    const float* Wq  = (const float*)d_in[7];
    const float* Wn  = (const float*)d_in[8];
    const float* bnp = (const float*)d_in[9];
    float* out = (float*)d_out;

    float* ws  = (float*)d_ws;
    float* mem = ws;  ws += (size_t)Bn * Sn * Dn;
    float* ast = ws;  ws += (size_t)Bn * Sn;
    float* en1 = ws;  ws += (size_t)Bn * Sn;
    float* en2 = ws;  ws += (size_t)Bn * Sn;
    float* aw  = ws;  ws += (size_t)Bn * Sn;
    float* qs  = ws;  ws += (size_t)Bn * Dn;
    float* qwq = ws;  ws += (size_t)Bn * Dn;
    float* rp  = ws;  ws += Bn;
    float* ap  = ws;  ws += Bn;
    float* lcn = ws;  ws += Bn;
    float* selp = ws; ws += (size_t)Bn * 4 * Dn;

    k_init<<<(Bn * Sn * Dn / 4) / 256, 256, 0, stream>>>((const v4f*)e, (v4f*)mem, ast);
    k_en<<<(Bn * Sn) / 8, 256, 0, stream>>>((const v4f*)e, (const v4f*)Wn, en1, en2);

    for (int t = 0; t < Tn; t++) {
        int kk = t + 1; if (kk > HIST) kk = HIST;
        // softmax over decay1 for valid lags (host; weights baked into kernel args)
        float tmp[HIST];
        double sum = 0.0;
        const float maxv = 1.0f; // decay1[19] = 1.0 is always valid
        for (int i = 0; i < HIST; i++) {
            if (i >= HIST - kk) { tmp[i] = expf((i + 1) / 20.0f - maxv); sum += tmp[i]; }
            else tmp[i] = 0.f;
        }
        LagW lw;
        for (int j = 0; j < HIST; j++) lw.w[j] = (float)(tmp[HIST - 1 - j] / sum);
        int jmax = (t < HIST - 1) ? t : HIST - 1;

        k_step_small<<<Bn, 256, 0, stream>>>(at, q, (const v4f*)Wr, br, (const v4f*)Wa, ba,
                                             en2, lw, t, jmax, aw, rp, ap, lcn);
        k_query_s<<<(Bn * Dn / 4) / 256, 256, 0, stream>>>((const v4f*)q, lw, t, jmax,
                                                           (v4f*)qs);
        k_gemm_wmma<<<64, 128, 0, stream>>>(qs, Wq, qwq);
        dim3 g3(4, Bn);
        k_step_big<<<g3, 256, 0, stream>>>(at, (const v4f*)e, (v4f*)mem, ast,
                                           (const v4f*)qwq, aw, rp, ap, lcn, en1,
                                           bnp, t, (v4f*)selp);
        k_sel_final<<<Bn, 256, 0, stream>>>((const v4f*)selp, (v4f*)out, t);
    }
}